// Decoder_10797547782619
// MI455X (gfx1250) — compile-verified
//
#include <hip/hip_runtime.h>

typedef __attribute__((ext_vector_type(2))) float v2f;
typedef __attribute__((ext_vector_type(8))) float v8f;

#define N_NODES 50000
#define N_EDGES 1250000
#define CH 64
#define NTILES (N_NODES / 16)   // 3125, exact

// ---------------------------------------------------------------------------
// Kernel 1: zero the aggregation buffer (lives in d_out, same 50000x64 shape)
// ---------------------------------------------------------------------------
__global__ void zero_kernel(float4* __restrict__ out, int n4) {
    int i = blockIdx.x * blockDim.x + threadIdx.x;
    if (i < n4) out[i] = make_float4(0.f, 0.f, 0.f, 0.f);
}

// ---------------------------------------------------------------------------
// Kernel 2: edge gather + scatter-add.  z and agg are L2-resident (12.8 MB
// each vs 192 MB L2), so this is L2 atomic-throughput bound. 2 channels/thread.
// ---------------------------------------------------------------------------
__global__ void scatter_kernel(const float* __restrict__ z,
                               const int*   __restrict__ ei,
                               float*       __restrict__ agg) {
    unsigned idx = blockIdx.x * blockDim.x + threadIdx.x;   // up to 40M < 2^31
    unsigned e = idx >> 5;
    if (e >= N_EDGES) return;
    int c   = (idx & 31) << 1;
    int src = ei[e];
    int dst = ei[N_EDGES + e];
    v2f v = *(const v2f*)(z + (size_t)src * CH + c);
    float* a = agg + (size_t)dst * CH + c;
    atomicAdd(a + 0, v.x);
    atomicAdd(a + 1, v.y);
}

// ---------------------------------------------------------------------------
// Kernel 3: fold the two linear layers:
//   Wrf = W_root @ W_lin,  Wnf = W_nei @ W_lin,  bf = b_conv @ W_lin + b_lin
// ---------------------------------------------------------------------------
__global__ void fuse_weights_kernel(const float* __restrict__ Wr,
                                    const float* __restrict__ Wn,
                                    const float* __restrict__ bc,
                                    const float* __restrict__ Wl,
                                    const float* __restrict__ bl,
                                    float* __restrict__ Wrf,
                                    float* __restrict__ Wnf,
                                    float* __restrict__ bf) {
    int t = threadIdx.x;                 // 0..255, each does 16 elements
    for (int i = 0; i < 16; ++i) {
        int o = t * 16 + i;              // 0..4095
        int r = o >> 6, c = o & 63;
        float s1 = 0.f, s2 = 0.f;
        for (int k = 0; k < 64; ++k) {
            float wl = Wl[k * 64 + c];
            s1 = __builtin_fmaf(Wr[r * 64 + k], wl, s1);
            s2 = __builtin_fmaf(Wn[r * 64 + k], wl, s2);
        }
        Wrf[o] = s1;
        Wnf[o] = s2;
    }
    if (t < 64) {
        float s = bl[t];
        for (int k = 0; k < 64; ++k) s = __builtin_fmaf(bc[k], Wl[k * 64 + t], s);
        bf[t] = s;
    }
}

// ---------------------------------------------------------------------------
// Kernel 4: fused GEMM, out = z @ Wrf + agg @ Wnf + bf, in-place over agg.
// One wave per 16-row node tile; V_WMMA_F32_16X16X4_F32 accumulation.
//
// A layout (16x4 f32, ISA 7.12.2): lane L: m=L%16, h=L/16 holds A[m][2h],A[m][2h+1]
// B layout (4x16 f32, mirrored):   lane L: n=L%16, h=L/16 holds B[2h][n],B[2h+1][n]
// C/D layout (16x16 f32):          lane L: n=L%16; VGPR r holds D[r+8*(L/16)][n]
//
// Weights are stored in LDS *pair-interleaved*: sW[(j*64+n)*2 + {0,1}] =
// W[2j][n], W[2j+1][n], so each lane's B fragment is one aligned ds_load_b64
// at (per-lane base + compile-time immediate) -> no register reassembly.
// ---------------------------------------------------------------------------
__global__ void __launch_bounds__(256)
gemm_kernel(const float* __restrict__ z,
            const float* __restrict__ Wrf,
            const float* __restrict__ Wnf,
            const float* __restrict__ bf,
            float* __restrict__ out /* agg on entry, result on exit */) {
    __shared__ float sWr[64 * 64];   // pair-interleaved
    __shared__ float sWn[64 * 64];   // pair-interleaved
    __shared__ float sb[64];

    int t = threadIdx.x;
    // Cooperative swizzling load: thread t handles row-pair j = t/8,
    // columns seg..seg+7 where seg = (t%8)*8.
    {
        int j   = t >> 3;            // 0..31  (rows 2j, 2j+1)
        int seg = (t & 7) * 8;       // 0..56
        const float* r0 = Wrf + (2 * j) * 64 + seg;
        const float* r1 = r0 + 64;
        float4 a0 = *(const float4*)(r0);
        float4 a1 = *(const float4*)(r0 + 4);
        float4 b0 = *(const float4*)(r1);
        float4 b1 = *(const float4*)(r1 + 4);
        float2* d = (float2*)(sWr + (j * 64 + seg) * 2);
        d[0] = make_float2(a0.x, b0.x); d[1] = make_float2(a0.y, b0.y);
        d[2] = make_float2(a0.z, b0.z); d[3] = make_float2(a0.w, b0.w);
        d[4] = make_float2(a1.x, b1.x); d[5] = make_float2(a1.y, b1.y);
        d[6] = make_float2(a1.z, b1.z); d[7] = make_float2(a1.w, b1.w);

        const float* s0 = Wnf + (2 * j) * 64 + seg;
        const float* s1 = s0 + 64;
        float4 c0 = *(const float4*)(s0);
        float4 c1 = *(const float4*)(s0 + 4);
        float4 e0 = *(const float4*)(s1);
        float4 e1 = *(const float4*)(s1 + 4);
        float2* g = (float2*)(sWn + (j * 64 + seg) * 2);
        g[0] = make_float2(c0.x, e0.x); g[1] = make_float2(c0.y, e0.y);
        g[2] = make_float2(c0.z, e0.z); g[3] = make_float2(c0.w, e0.w);
        g[4] = make_float2(c1.x, e1.x); g[5] = make_float2(c1.y, e1.y);
        g[6] = make_float2(c1.z, e1.z); g[7] = make_float2(c1.w, e1.w);
    }
    if (t < 64) sb[t] = bf[t];
    __syncthreads();

    int wave = t >> 5;                   // wave32: 8 waves per block
    int lane = t & 31;
    int tile = blockIdx.x * 8 + wave;    // uniform per wave
    if (tile >= NTILES) return;

    int n = lane & 15;                   // A row index == B/C/D column index
    int h = lane >> 4;
    size_t rowbase = ((size_t)tile * 16 + n) * CH;

    // Preload ALL A-fragments for this tile (z and agg) before any store,
    // so the in-place overwrite of agg is safe.
    v2f az[16], ag[16];
#pragma unroll
    for (int kk = 0; kk < 16; ++kk) {
        int col = kk * 4 + h * 2;
        az[kk] = *(const v2f*)(z   + rowbase + col);
        ag[kk] = *(const v2f*)(out + rowbase + col);
    }

    // Per-lane LDS base in v2f units; fragment (kk, nt) is at +kk*128 + nt*16.
    const v2f* pR = ((const v2f*)sWr) + (h * 64 + n);
    const v2f* pN = ((const v2f*)sWn) + (h * 64 + n);

#pragma unroll
    for (int nt = 0; nt < 4; ++nt) {
        float bias = sb[nt * 16 + n];
        v8f acc;
#pragma unroll
        for (int r = 0; r < 8; ++r) acc[r] = bias;

#pragma unroll
        for (int kk = 0; kk < 16; ++kk) {
            v2f br = pR[kk * 128 + nt * 16];   // ds_load_b64, imm offset
            v2f bn = pN[kk * 128 + nt * 16];   // ds_load_b64, imm offset
            acc = __builtin_amdgcn_wmma_f32_16x16x4_f32(
                false, az[kk], false, br, (short)0, acc, false, false);
            acc = __builtin_amdgcn_wmma_f32_16x16x4_f32(
                false, ag[kk], false, bn, (short)0, acc, false, false);
        }

#pragma unroll
        for (int r = 0; r < 8; ++r) {
            size_t orow = (size_t)tile * 16 + r + 8 * h;
            out[orow * CH + nt * 16 + n] = acc[r];
        }
    }
}

// ---------------------------------------------------------------------------
extern "C" void kernel_launch(void* const* d_in, const int* in_sizes, int n_in,
                              void* d_out, int out_size, void* d_ws, size_t ws_size,
                              hipStream_t stream) {
    const float* z  = (const float*)d_in[0];
    const int*   ei = (const int*)  d_in[1];   // [2, E] row-major int32
    const float* Wr = (const float*)d_in[2];
    const float* Wn = (const float*)d_in[3];
    const float* bc = (const float*)d_in[4];
    const float* Wl = (const float*)d_in[5];
    const float* bl = (const float*)d_in[6];
    float* out = (float*)d_out;

    float* ws  = (float*)d_ws;
    float* Wrf = ws;            // 4096 f32
    float* Wnf = ws + 4096;     // 4096 f32
    float* bf  = ws + 8192;     // 64 f32   (~33 KB total workspace)

    // 1) zero agg buffer (reuse d_out: identical 50000x64 shape)
    int n4 = N_NODES * CH / 4;
    zero_kernel<<<(n4 + 255) / 256, 256, 0, stream>>>((float4*)out, n4);

    // 2) fold weights
    fuse_weights_kernel<<<1, 256, 0, stream>>>(Wr, Wn, bc, Wl, bl, Wrf, Wnf, bf);

    // 3) edge scatter-add into agg (L2-resident atomics)
    unsigned sthreads = (unsigned)N_EDGES * 32u;   // 2 channels per thread
    scatter_kernel<<<(sthreads + 255) / 256, 256, 0, stream>>>(z, ei, out);

    // 4) fused WMMA GEMM, in-place agg -> out
    gemm_kernel<<<(NTILES + 7) / 8, 256, 0, stream>>>(z, Wrf, Wnf, bf, out);
}